// GraphSAGE_50371376447710
// MI455X (gfx1250) — compile-verified
//
#include <hip/hip_runtime.h>

typedef __attribute__((ext_vector_type(2))) float v2f;
typedef __attribute__((ext_vector_type(8))) float v8f;

#define N_NODES 100000
#define N_EDGES 1000000
#define D 64

// ---------------------------------------------------------------------------
// Zero-fill (float4 grid-stride). Traffic is tiny (~50 MB) vs 23.3 TB/s HBM.
// ---------------------------------------------------------------------------
__global__ void __launch_bounds__(256) zero_f4(float4* __restrict__ p, int n4) {
    int i      = blockIdx.x * blockDim.x + threadIdx.x;
    int stride = gridDim.x * blockDim.x;
    float4 z = {0.f, 0.f, 0.f, 0.f};
    for (; i < n4; i += stride) p[i] = z;
}

// ---------------------------------------------------------------------------
// Edge scatter: one wave32 per edge. Each lane carries 2 of the 64 features.
// 2x global_atomic_add_f32 per lane; agg (25.6 MB) is L2-resident on MI455X
// (192 MB L2) so the 64M atomics/layer resolve at L2 rate.
// ---------------------------------------------------------------------------
__global__ void __launch_bounds__(256)
sage_scatter(const float* __restrict__ feat,   // [N, 64] source features
             const int*   __restrict__ ei,     // [2, E] (src row, dst row)
             float*       __restrict__ agg,    // [N, 64] scatter-sum target
             float*       __restrict__ cnt,    // [N] degree (pass 1 only)
             int n_edges, int do_count) {
    const int gtid = blockIdx.x * blockDim.x + threadIdx.x;
    const int e    = gtid >> 5;
    const int lane = threadIdx.x & 31;
    if (e >= n_edges) return;

    const int src = ei[e];
    const int dst = ei[n_edges + e];

    const float2 v = *(const float2*)(feat + (long)src * D + lane * 2);
    float* ap = agg + (long)dst * D + lane * 2;
    atomicAdd(ap,     v.x);
    atomicAdd(ap + 1, v.y);
    if (do_count && lane == 0) atomicAdd(cnt + dst, 1.0f);
}

// ---------------------------------------------------------------------------
// Fused SAGE linear: out = relu?( (agg/max(cnt,1)) @ Wl^T + bl + xin @ Wr^T )
// One wave32 per 16-row node tile, producing 16x64 via V_WMMA_F32_16X16X4_F32.
// A layout (16x4 f32): lane<16 holds (M=lane, K=k,k+1); lane>=16 (M, K=k+2,k+3)
//   -> one contiguous float2 load per lane per K-step.
// B layout (4x16): mirrored -> float2 load of W[n][k..k+1], n = 16c + (lane&15).
// C/D layout: vgpr r -> row (r + 8*(lane>>4)), col = 16c + (lane&15).
// ---------------------------------------------------------------------------
__global__ void __launch_bounds__(256)
sage_gemm(const float* __restrict__ agg,   // [N,64] neighbor sums
          const float* __restrict__ cnt,   // [N] degrees
          const float* __restrict__ xin,   // [N,64] self features
          const float* __restrict__ Wl,    // [64,64] (out, in)
          const float* __restrict__ bl,    // [64]
          const float* __restrict__ Wr,    // [64,64] (out, in)
          float*       __restrict__ out,   // [N,64]
          int n_nodes, int relu) {
    const int lane   = threadIdx.x & 31;
    const int wave   = threadIdx.x >> 5;
    const int tile   = blockIdx.x * (blockDim.x >> 5) + wave;
    const int ntiles = n_nodes >> 4;          // wave-uniform guard: EXEC stays ~0
    if (tile >= ntiles) return;

    const int row0 = tile << 4;
    const int half = lane >> 4;               // 0: K lo pair, 1: K hi pair
    const int mrow = lane & 15;

    const float invc = 1.0f / fmaxf(cnt[row0 + mrow], 1.0f);

    const float* arow = agg + (long)(row0 + mrow) * D + 2 * half;
    const float* xrow = xin + (long)(row0 + mrow) * D + 2 * half;

    v8f acc[4] = {v8f{}, v8f{}, v8f{}, v8f{}};

    for (int pass = 0; pass < 2; ++pass) {
        const float* A = pass ? xrow : arow;
        const float* W = pass ? Wr : Wl;
        const float* w0 = W + (0 * 16 + mrow) * D + 2 * half;
        const float* w1 = W + (1 * 16 + mrow) * D + 2 * half;
        const float* w2 = W + (2 * 16 + mrow) * D + 2 * half;
        const float* w3 = W + (3 * 16 + mrow) * D + 2 * half;
#pragma unroll
        for (int k = 0; k < D; k += 4) {
            v2f a = *(const v2f*)(A + k);
            if (pass == 0) a *= invc;         // fold the neighbor mean into A
            v2f b0 = *(const v2f*)(w0 + k);
            v2f b1 = *(const v2f*)(w1 + k);
            v2f b2 = *(const v2f*)(w2 + k);
            v2f b3 = *(const v2f*)(w3 + k);
            acc[0] = __builtin_amdgcn_wmma_f32_16x16x4_f32(false, a, false, b0,
                                                           (short)0, acc[0], false, false);
            acc[1] = __builtin_amdgcn_wmma_f32_16x16x4_f32(false, a, false, b1,
                                                           (short)0, acc[1], false, false);
            acc[2] = __builtin_amdgcn_wmma_f32_16x16x4_f32(false, a, false, b2,
                                                           (short)0, acc[2], false, false);
            acc[3] = __builtin_amdgcn_wmma_f32_16x16x4_f32(false, a, false, b3,
                                                           (short)0, acc[3], false, false);
        }
    }

    // epilogue: bias (+ReLU) and store. Lanes 0-15 write row r, 16-31 row r+8:
    // two contiguous 64B segments per store -> fully coalesced.
#pragma unroll
    for (int c = 0; c < 4; ++c) {
        const float bv = bl[c * 16 + mrow];
        float* obase = out + (long)(row0 + 8 * half) * D + c * 16 + mrow;
#pragma unroll
        for (int r = 0; r < 8; ++r) {
            float v = acc[c][r] + bv;
            if (relu) v = fmaxf(v, 0.0f);
            obase[(long)r * D] = v;
        }
    }
}

// ---------------------------------------------------------------------------
// Host-side launcher (graph-capture safe: no malloc/sync, stream-only).
// ---------------------------------------------------------------------------
extern "C" void kernel_launch(void* const* d_in, const int* in_sizes, int n_in,
                              void* d_out, int out_size, void* d_ws, size_t ws_size,
                              hipStream_t stream) {
    const float* x   = (const float*)d_in[0];
    const int*   ei  = (const int*)  d_in[1];
    const float* W1l = (const float*)d_in[2];
    const float* b1l = (const float*)d_in[3];
    const float* W1r = (const float*)d_in[4];
    const float* W2l = (const float*)d_in[5];
    const float* b2l = (const float*)d_in[6];
    const float* W2r = (const float*)d_in[7];
    float* out = (float*)d_out;

    // workspace layout: agg[N*64] | cnt[N] | h[N*64]  (~51.6 MB)
    float* agg = (float*)d_ws;
    float* cnt = agg + (size_t)N_NODES * D;
    float* h   = cnt + N_NODES;

    const int scatter_blocks = (N_EDGES * 32) / 256;          // one wave per edge
    const int gemm_blocks    = ((N_NODES / 16) + 7) / 8;      // 8 waves per block

    // ---- layer 1 ----
    zero_f4<<<2048, 256, 0, stream>>>((float4*)agg, (N_NODES * (D + 1)) / 4); // agg+cnt
    sage_scatter<<<scatter_blocks, 256, 0, stream>>>(x, ei, agg, cnt, N_EDGES, 1);
    sage_gemm<<<gemm_blocks, 256, 0, stream>>>(agg, cnt, x, W1l, b1l, W1r, h,
                                               N_NODES, 1);
    // ---- layer 2 ----
    zero_f4<<<2048, 256, 0, stream>>>((float4*)agg, (N_NODES * D) / 4);
    sage_scatter<<<scatter_blocks, 256, 0, stream>>>(h, ei, agg, cnt, N_EDGES, 0);
    sage_gemm<<<gemm_blocks, 256, 0, stream>>>(agg, cnt, h, W2l, b2l, W2r, out,
                                               N_NODES, 0);
}